// StateEstimator_20169166422533
// MI455X (gfx1250) — compile-verified
//
#include <hip/hip_runtime.h>

typedef __attribute__((ext_vector_type(2))) float v2f;
typedef __attribute__((ext_vector_type(8))) float v8f;

namespace {
constexpr int kT = 256;          // timesteps
constexpr int kF = 9;            // input features
constexpr int kU = 32;           // units per population (spiking / integrator)
constexpr int kO = 9;            // output features per step
constexpr int kOutRow = kT * kO; // 2304 floats per batch row

// D = A(16x4,f32) * B(4x16,f32) + C(16x16,f32)
__device__ __forceinline__ v8f wmma4(v2f a, v2f b, v8f c) {
  return __builtin_amdgcn_wmma_f32_16x16x4_f32(false, a, false, b, (short)0, c,
                                               false, false);
}

// Build one operand VGPR of the s-matrix for GEMM2 from state registers.
// State layout (D of GEMM1): value(unit,batch) lives in lane (batch + 16*hs)
// of state vgpr (unit&7). Needed layout (A of GEMM2):
//   lanes 0-15  -> s[batch=lane,    unit_lo]   (source half = hs)
//   lanes 16-31 -> s[batch=lane-16, unit_lo+2] (source half = hs)
// hs is compile-time per chunk, so one half of the wave always holds its value
// in place -> a single ds_bpermute + cndmask per slot.
__device__ __forceinline__ float gpair0(float lo, float hi) {
  // hs == 0: lanes<16 take lo in place; lanes>=16 fetch hi[lane-16]
  const float b = __shfl(hi, threadIdx.x & 15, 32);
  return (threadIdx.x & 16) ? b : lo;
}
__device__ __forceinline__ float gpair1(float lo, float hi) {
  // hs == 1: lanes<16 fetch lo[lane+16]; lanes>=16 take hi in place
  const float a = __shfl(lo, (threadIdx.x & 15) | 16, 32);
  return (threadIdx.x & 16) ? hi : a;
}

// Load the three 4x16 B-chunks of x_t^T, branchlessly (read-once -> NT).
// Per lane: features {0,1,4,5} (half 0) / {2,3,6,7} (half 1) -> two B64 loads;
// feature 8 loaded by ALL lanes (same cacheline) and masked to the low half;
// features 9..11 are K-padding -> compile-time zero, no load.
__device__ __forceinline__ void loadBx(const float* __restrict__ xrow, int t,
                                       int half, v2f Bx[3]) {
  const float* p = xrow + t * kF;
#pragma unroll
  for (int kc = 0; kc < 2; ++kc)
#pragma unroll
    for (int i = 0; i < 2; ++i)
      Bx[kc][i] = __builtin_nontemporal_load(p + 4 * kc + i + 2 * half);
  const float x8 = __builtin_nontemporal_load(p + 8);
  Bx[2][0] = half ? 0.0f : x8; // K=8 valid only in low half (K = i + 2*half)
  Bx[2][1] = 0.0f;             // K=9/11: padding
}
} // namespace

__global__ __launch_bounds__(256) void snn_lif_wmma_kernel(
    const float* __restrict__ X, const float* __restrict__ W1,
    const float* __restrict__ b1, const float* __restrict__ betas,
    const float* __restrict__ thrS, const float* __restrict__ thrI,
    const float* __restrict__ W2, const float* __restrict__ b2,
    float* __restrict__ Out, int nTiles) {
  const int wave = blockIdx.x * (blockDim.x >> 5) + (threadIdx.x >> 5);
  if (wave >= nTiles) return; // whole-wave exit keeps EXEC all-ones for WMMA
  const int lane = threadIdx.x & 31;
  const int col = lane & 15; // M/N index inside a 16-wide tile
  const int half = lane >> 4;
  const int baseB = wave * 16;

  // ---------------- constant WMMA operands (VGPR-resident) ----------------
  // A-chunks of W1[32,9] for GEMM1 (h^T = W1 * x^T): M=unit row, K=feature.
  // 32-bit chunk layout: K = vgpr_idx + 2*half, M = col.
  v2f Aw1[2][3];
#pragma unroll
  for (int tile = 0; tile < 2; ++tile)
#pragma unroll
    for (int kc = 0; kc < 3; ++kc)
#pragma unroll
      for (int i = 0; i < 2; ++i) {
        const int k = 4 * kc + i + 2 * half;
        Aw1[tile][kc][i] = (k < kF) ? W1[(16 * tile + col) * kF + k] : 0.0f;
      }

  // B-chunks of W2^T[64,16] for GEMM2 (out = s * W2^T): K=unit, N=out chan.
  v2f Bw2[16];
#pragma unroll
  for (int kc = 0; kc < 16; ++kc)
#pragma unroll
    for (int i = 0; i < 2; ++i) {
      const int u = 4 * kc + i + 2 * half;
      Bw2[kc][i] = (col < kO) ? W2[col * (2 * kU) + u] : 0.0f;
    }

  // Per-slot LIF parameters in the state layout: unit = 16*tile + v + 8*half.
  float hb[2][8], bet[2][8], tS[2][8], tI[2][8];
#pragma unroll
  for (int tile = 0; tile < 2; ++tile)
#pragma unroll
    for (int v = 0; v < 8; ++v) {
      const int u = 16 * tile + v + 8 * half;
      hb[tile][v] = b1[u];
      bet[tile][v] = betas[u];
      tS[tile][v] = thrS[u];
      tI[tile][v] = thrI[u];
    }
  const float cb = (col < kO) ? b2[col] : 0.0f; // bias for GEMM2 accumulator

  // Membrane state, zero-initialized.
  float m1[2][8], m2[2][8];
#pragma unroll
  for (int tile = 0; tile < 2; ++tile)
#pragma unroll
    for (int v = 0; v < 8; ++v) {
      m1[tile][v] = 0.0f;
      m2[tile][v] = 0.0f;
    }

  const float* __restrict__ xrow =
      X + (size_t)(baseB + col) * (kT * kF); // this lane's batch row
  float* __restrict__ orow =
      Out + (size_t)(baseB + 8 * half) * kOutRow + col;

  // Software pipeline: x_t loads issued one iteration ahead.
  v2f Bx[3];
  loadBx(xrow, 0, half, Bx);

  for (int t = 0; t < kT; ++t) {
    // -------- GEMM1 + LIF state update --------
    float s1[2][8], s2[2][8];
#pragma unroll
    for (int tile = 0; tile < 2; ++tile) {
      v8f c;
#pragma unroll
      for (int v = 0; v < 8; ++v) c[v] = hb[tile][v]; // h starts at b1
      c = wmma4(Aw1[tile][0], Bx[0], c);
      c = wmma4(Aw1[tile][1], Bx[1], c);
      c = wmma4(Aw1[tile][2], Bx[2], c);
#pragma unroll
      for (int v = 0; v < 8; ++v) {
        const float h = c[v];
        // spiking LIF: reset-by-subtract from *previous* membrane
        const float mo1 = m1[tile][v];
        const float r1 = (mo1 > tS[tile][v]) ? tS[tile][v] : 0.0f;
        const float mn1 = __builtin_fmaf(bet[tile][v], mo1, h) - r1;
        m1[tile][v] = mn1;
        s1[tile][v] = (mn1 > tS[tile][v]) ? 1.0f : 0.0f;
        // perfect integrator (beta = 1)
        const float mo2 = m2[tile][v];
        const float r2 = (mo2 > tI[tile][v]) ? tI[tile][v] : 0.0f;
        const float mn2 = mo2 + h - r2;
        m2[tile][v] = mn2;
        s2[tile][v] = (mn2 > tI[tile][v]) ? 1.0f : 0.0f;
      }
    }

    // Prefetch next timestep's x chunks; first use is ~150 instructions away
    // (GEMM2 + stores + next iteration's top), hiding the L2/HBM latency.
    v2f BxN[3];
    {
      const int tn = (t + 1 < kT) ? (t + 1) : t; // clamped: no OOB on last row
      loadBx(xrow, tn, half, BxN);
    }

    // -------- GEMM2: out[16 batch, 16] = [s1|s2] * W2^T + b2 --------
    v8f co;
#pragma unroll
    for (int v = 0; v < 8; ++v) co[v] = cb;
#pragma unroll
    for (int kc = 0; kc < 16; ++kc) {
      const int uu = (4 * kc) & 15;
      const int hs = uu >> 3; // compile-time after unroll
      const int vs = uu & 7;
      float a0lo, a0hi, a1lo, a1hi;
      if (kc < 4) {
        a0lo = s1[0][vs]; a0hi = s1[0][vs + 2];
        a1lo = s1[0][vs + 1]; a1hi = s1[0][vs + 3];
      } else if (kc < 8) {
        a0lo = s1[1][vs]; a0hi = s1[1][vs + 2];
        a1lo = s1[1][vs + 1]; a1hi = s1[1][vs + 3];
      } else if (kc < 12) {
        a0lo = s2[0][vs]; a0hi = s2[0][vs + 2];
        a1lo = s2[0][vs + 1]; a1hi = s2[0][vs + 3];
      } else {
        a0lo = s2[1][vs]; a0hi = s2[1][vs + 2];
        a1lo = s2[1][vs + 1]; a1hi = s2[1][vs + 3];
      }
      v2f As;
      if (hs) {
        As[0] = gpair1(a0lo, a0hi);
        As[1] = gpair1(a1lo, a1hi);
      } else {
        As[0] = gpair0(a0lo, a0hi);
        As[1] = gpair0(a1lo, a1hi);
      }
      co = wmma4(As, Bw2[kc], co);
    }

    // -------- stream out[b, t*9 + j] (write-once -> non-temporal) --------
    // D layout: batch = baseB + v + 8*half, j = col (valid j < 9).
    if (col < kO) {
      float* p = orow + (size_t)t * kO;
#pragma unroll
      for (int v = 0; v < 8; ++v)
        __builtin_nontemporal_store(co[v], p + (size_t)v * kOutRow);
    }

    Bx[0] = BxN[0];
    Bx[1] = BxN[1];
    Bx[2] = BxN[2];
  }
}

extern "C" void kernel_launch(void* const* d_in, const int* in_sizes, int n_in,
                              void* d_out, int out_size, void* d_ws,
                              size_t ws_size, hipStream_t stream) {
  (void)n_in; (void)out_size; (void)d_ws; (void)ws_size;
  const float* X    = (const float*)d_in[0];
  const float* W1   = (const float*)d_in[1];
  const float* b1   = (const float*)d_in[2];
  const float* bet  = (const float*)d_in[3];
  const float* thrS = (const float*)d_in[4];
  const float* thrI = (const float*)d_in[5];
  const float* W2   = (const float*)d_in[6];
  const float* b2   = (const float*)d_in[7];
  float* Out = (float*)d_out;

  const int B = in_sizes[0] / (kT * kF); // 4096
  const int nTiles = B / 16;             // 256 waves, one 16-batch tile each
  const int threads = 256;               // 8 waves per block
  const int wavesPerBlock = threads / 32;
  const int blocks = (nTiles + wavesPerBlock - 1) / wavesPerBlock;

  snn_lif_wmma_kernel<<<blocks, threads, 0, stream>>>(
      X, W1, b1, bet, thrS, thrI, W2, b2, Out, nTiles);
}